// TensorDLT_16527034155638
// MI455X (gfx1250) — compile-verified
//
#include <hip/hip_runtime.h>
#include <hip/hip_bf16.h>
#include <stdint.h>

// TensorDLT: closed-form 4-point homography (Heckbert square->quad), memory-bound.
// Input staged global->LDS via gfx1250 async tensor-path loads; output drained
// LDS->global via async b128 stores for coalescing of the 36B/thread H matrices.

#define TDL_BLOCK 256

__global__ __launch_bounds__(TDL_BLOCK) void
tensor_dlt_kernel(const float* __restrict__ offset, float* __restrict__ out, int B)
{
    __shared__ __align__(16) float s_in[TDL_BLOCK * 8];    // 8 KB staged input
    __shared__ __align__(16) float s_out[TDL_BLOCK * 9];   // 9 KB staged output

    const int tid = threadIdx.x;
    const int g   = blockIdx.x * TDL_BLOCK + tid;

    // ---- stage input: 32 B per lane, global -> LDS (async, no VGPR round trip)
    if (g < B) {
        uint32_t lin = (uint32_t)(uintptr_t)(&s_in[tid * 8]);   // LDS byte addr = low 32 bits of generic ptr
        const float* src = offset + (size_t)g * 8;
        asm volatile(
            "global_load_async_to_lds_b128 %0, %1, off\n\t"
            "global_load_async_to_lds_b128 %0, %1, off offset:16"
            :: "v"(lin), "v"(src) : "memory");
    }
    asm volatile("s_wait_asynccnt 0" ::: "memory");

    if (g < B) {
        const float* o = &s_in[tid * 8];
        const float o0 = o[0], o1 = o[1], o2 = o[2], o3 = o[3];
        const float o4 = o[4], o5 = o[5], o6 = o[6], o7 = o[7];

        // destination corners (src corners are (0,0),(127,0),(127,127),(0,127))
        const float x0 =          32.0f * o0, y0 =          32.0f * o1;
        const float x1 = 127.0f + 32.0f * o2, y1 =          32.0f * o3;
        const float x2 = 127.0f + 32.0f * o4, y2 = 127.0f + 32.0f * o5;
        const float x3 =          32.0f * o6, y3 = 127.0f + 32.0f * o7;

        // Heckbert unit-square -> quad homography (degenerates cleanly to affine)
        const float sx  = x0 - x1 + x2 - x3;
        const float sy  = y0 - y1 + y2 - y3;
        const float dx1 = x1 - x2, dy1 = y1 - y2;
        const float dx2 = x3 - x2, dy2 = y3 - y2;
        const float det = dx1 * dy2 - dx2 * dy1;
        const float inv = 1.0f / det;
        const float gg  = (sx * dy2 - dx2 * sy) * inv;
        const float hh  = (dx1 * sy - sx * dy1) * inv;
        const float a   = x1 - x0 + gg * x1;
        const float b   = x3 - x0 + hh * x3;
        const float d   = y1 - y0 + gg * y1;
        const float e   = y3 - y0 + hh * y3;

        // compose with scale(1/127) on source coords; H[2][2] stays 1 == DLT h9
        const float s = 1.0f / 127.0f;
        float* w = &s_out[tid * 9];
        w[0] = a  * s;  w[1] = b  * s;  w[2] = x0;
        w[3] = d  * s;  w[4] = e  * s;  w[5] = y0;
        w[6] = gg * s;  w[7] = hh * s;  w[8] = 1.0f;
    }

    __syncthreads();

    // ---- drain output: coalesced async LDS -> global
    const int base_dw  = blockIdx.x * (TDL_BLOCK * 9);   // dword offset of this block's tile
    const int total_dw = B * 9;

    if (base_dw + TDL_BLOCK * 9 <= total_dw) {
        // full tile: 576 x b128 chunks, 16B-aligned both sides, no guards needed
#pragma unroll
        for (int k = 0; k < 3; ++k) {
            const int chunk = k * TDL_BLOCK + tid;
            if (chunk < (TDL_BLOCK * 9) / 4) {           // 576; folds for k=0,1
                uint32_t lofs = (uint32_t)(uintptr_t)(&s_out[chunk * 4]);
                float* gp = out + base_dw + chunk * 4;
                asm volatile("global_store_async_from_lds_b128 %0, %1, off"
                             :: "v"(gp), "v"(lofs) : "memory");
            }
        }
    } else {
        // tail tile: guarded b32 drain (never writes past B*9)
#pragma unroll
        for (int k = 0; k < 9; ++k) {
            const int idx  = k * TDL_BLOCK + tid;
            const int gidx = base_dw + idx;
            if (gidx < total_dw) {
                uint32_t lofs = (uint32_t)(uintptr_t)(&s_out[idx]);
                float* gp = out + gidx;
                asm volatile("global_store_async_from_lds_b32 %0, %1, off"
                             :: "v"(gp), "v"(lofs) : "memory");
            }
        }
    }
    asm volatile("s_wait_asynccnt 0" ::: "memory");
}

extern "C" void kernel_launch(void* const* d_in, const int* in_sizes, int n_in,
                              void* d_out, int out_size, void* d_ws, size_t ws_size,
                              hipStream_t stream)
{
    const float* offset = (const float*)d_in[0];   // [B,8] f32; d_in[1] (C_a) unused by the math
    float* out = (float*)d_out;                    // [B,3,3] f32
    const int B = in_sizes[0] / 8;
    if (B <= 0) return;
    const int grid = (B + TDL_BLOCK - 1) / TDL_BLOCK;
    tensor_dlt_kernel<<<grid, TDL_BLOCK, 0, stream>>>(offset, out, B);
}